// GCN_30966714204819
// MI455X (gfx1250) — compile-verified
//
#include <hip/hip_runtime.h>

typedef __attribute__((ext_vector_type(2))) float v2f;
typedef __attribute__((ext_vector_type(8))) float v8f;

#define NN 100000      // nodes
#define NG 512         // graphs

// ---------------- utility kernels ----------------

__global__ void k_zero(float* __restrict__ p, long long n) {
    long long i = blockIdx.x * (long long)blockDim.x + threadIdx.x;
    if (i < n) p[i] = 0.0f;
}

__global__ void k_fill1(float* __restrict__ p, int n) {
    int i = blockIdx.x * blockDim.x + threadIdx.x;
    if (i < n) p[i] = 1.0f;          // self-loop contributes 1 to every degree
}

__global__ void k_deg_accum(const int* __restrict__ dst, float* __restrict__ deg, int nE) {
    int e = blockIdx.x * blockDim.x + threadIdx.x;
    if (e < nE) atomicAdd(&deg[dst[e]], 1.0f);
}

__global__ void k_rsqrt(float* __restrict__ d, int n) {
    int i = blockIdx.x * blockDim.x + threadIdx.x;
    if (i < n) d[i] = rsqrtf(d[i]);  // deg >= 1 always (self-loops), no zero guard needed
}

// ---------------- WMMA f32 GEMM: C[M,N] = A[M,K] @ B[K,N] (+bias, relu) ----------------
// One wave computes one 16x16 output tile with V_WMMA_F32_16X16X4_F32.
// Requires M%16==0, N%16==0, K%4==0 (true for all call sites).
__global__ void k_wmma_gemm(const float* __restrict__ A, const float* __restrict__ B,
                            float* __restrict__ C, const float* __restrict__ bias,
                            int M, int N, int K, int relu) {
    int nTn    = N >> 4;
    int nTiles = (M >> 4) * nTn;
    int tile   = blockIdx.x * (blockDim.x >> 5) + (threadIdx.x >> 5);
    if (tile >= nTiles) return;      // whole-wave exit: EXEC all-1s for WMMA in live waves

    int tm   = tile / nTn;
    int tn   = tile - tm * nTn;
    int lane = threadIdx.x & 31;
    int half = lane >> 4;            // 0: K=k,k+1  1: K=k+2,k+3 (A layout, ISA 7.12.2)
    int l    = lane & 15;

    int arow = tm * 16 + l;          // A: M striped across lanes (both halves)
    int col  = tn * 16 + l;          // B/C/D: N striped across lanes (both halves)
    const float* __restrict__ Ar = A + (long long)arow * K;

    v8f c = {};
    for (int k = 0; k < K; k += 4) {
        int ka = k + half * 2;
        v2f a, b;
        a.x = Ar[ka];
        a.y = Ar[ka + 1];
        b.x = B[(long long)ka       * N + col];
        b.y = B[(long long)(ka + 1) * N + col];
        c = __builtin_amdgcn_wmma_f32_16x16x4_f32(
                /*neg_a=*/false, a, /*neg_b=*/false, b,
                /*c_mod=*/(short)0, c, /*reuse_a=*/false, /*reuse_b=*/false);
    }

    float bv = bias ? bias[col] : 0.0f;
#pragma unroll
    for (int i = 0; i < 8; i++) {    // VGPR i -> rows i (lanes 0-15) and i+8 (lanes 16-31)
        int row = tm * 16 + half * 8 + i;
        float v = c[i] + bv;
        if (relu) v = fmaxf(v, 0.0f);
        C[(long long)row * N + col] = v;
    }
}

// ---------------- edge scatter: agg[dst] += h[src] * dinv[src]*dinv[dst] ----------------
// One thread per (edge, feature); consecutive tids -> consecutive features -> coalesced.
__global__ void k_scatter(const float* __restrict__ h, const int* __restrict__ src,
                          const int* __restrict__ dst, const float* __restrict__ dinv,
                          float* __restrict__ agg, int nE, int logF) {
    long long tid   = blockIdx.x * (long long)blockDim.x + threadIdx.x;
    long long total = (long long)nE << logF;
    if (tid >= total) return;
    int F = 1 << logF;
    int e = (int)(tid >> logF);
    int f = (int)(tid & (F - 1));
    int s = src[e], d = dst[e];
    float nrm = dinv[s] * dinv[d];
    atomicAdd(&agg[(long long)d * F + f], h[(long long)s * F + f] * nrm);
}

// out = relu(agg + h*dinv^2 + bias)  (self-loop term; agg may alias out)
__global__ void k_selfloop_bias_relu(const float* __restrict__ h, const float* __restrict__ dinv,
                                     const float* __restrict__ agg, const float* __restrict__ bias,
                                     float* __restrict__ out, int n, int logF) {
    long long tid   = blockIdx.x * (long long)blockDim.x + threadIdx.x;
    long long total = (long long)n << logF;
    if (tid >= total) return;
    int F = 1 << logF;
    int i = (int)(tid >> logF);
    int f = (int)(tid & (F - 1));
    float dv = dinv[i];
    float v  = agg[tid] + h[tid] * dv * dv + bias[f];
    out[tid] = fmaxf(v, 0.0f);
}

// ---------------- pooling ----------------

__global__ void k_pool_accum(const float* __restrict__ x1, const float* __restrict__ x2,
                             const int* __restrict__ batch, float* __restrict__ sums,
                             float* __restrict__ cnt, int n) {
    long long tid = blockIdx.x * (long long)blockDim.x + threadIdx.x;
    if (tid >= (long long)n * 96) return;
    int i = (int)(tid / 96);
    int f = (int)(tid - (long long)i * 96);
    int g = batch[i];
    float v = (f < 32) ? x1[(long long)i * 32 + f] : x2[(long long)i * 64 + (f - 32)];
    atomicAdd(&sums[g * 96 + f], v);
    if (f == 0) atomicAdd(&cnt[g], 1.0f);
}

__global__ void k_pool_final(float* __restrict__ sums, const float* __restrict__ cnt, int g96) {
    int i = blockIdx.x * blockDim.x + threadIdx.x;
    if (i < g96) sums[i] /= fmaxf(cnt[i / 96], 1.0f);
}

// out[g] = hidden[g,:] . fW2 + fb2
__global__ void k_head2(const float* __restrict__ hidden, const float* __restrict__ fW2,
                        const float* __restrict__ fb2, float* __restrict__ out, int G) {
    int g = blockIdx.x * blockDim.x + threadIdx.x;
    if (g >= G) return;
    float acc = fb2[0];
#pragma unroll 8
    for (int k = 0; k < 128; k++) acc += hidden[(long long)g * 128 + k] * fW2[k];
    out[g] = acc;
}

// ---------------- launch ----------------

extern "C" void kernel_launch(void* const* d_in, const int* in_sizes, int n_in,
                              void* d_out, int out_size, void* d_ws, size_t ws_size,
                              hipStream_t stream) {
    const float* x     = (const float*)d_in[0];
    const int*   ei    = (const int*)  d_in[1];
    const int*   batch = (const int*)  d_in[2];
    const float* W1    = (const float*)d_in[3];
    const float* b1    = (const float*)d_in[4];
    const float* W2    = (const float*)d_in[5];
    const float* b2    = (const float*)d_in[6];
    const float* fW1   = (const float*)d_in[7];
    const float* fb1   = (const float*)d_in[8];
    const float* fW2   = (const float*)d_in[9];
    const float* fb2   = (const float*)d_in[10];
    float* out = (float*)d_out;

    const int E   = in_sizes[1] / 2;
    const int* src = ei;
    const int* dst = ei + E;

    // workspace layout (floats); x1 aliases agg1, x2 aliases agg2, pooled aliases sums
    float* ws     = (float*)d_ws;
    float* dinv   = ws;                            // NN
    float* h1     = dinv + NN;                     // NN*32
    float* x1     = h1   + (long long)NN * 32;     // NN*32  (agg1 -> x1 in place)
    float* h2     = x1   + (long long)NN * 32;     // NN*64
    float* x2     = h2   + (long long)NN * 64;     // NN*64  (agg2 -> x2 in place)
    float* sums   = x2   + (long long)NN * 64;     // NG*96  (-> pooled in place)
    float* cnt    = sums + NG * 96;                // NG
    float* hidden = cnt  + NG;                     // NG*128

    const int T = 256;
    const int WPB = T / 32;  // waves per block in GEMM
    auto blocks = [](long long n, int t) { return (unsigned)((n + t - 1) / t); };

    // --- degree / normalization ---
    k_fill1   <<<blocks(NN, T), T, 0, stream>>>(dinv, NN);
    k_deg_accum<<<blocks(E, T),  T, 0, stream>>>(dst, dinv, E);
    k_rsqrt   <<<blocks(NN, T), T, 0, stream>>>(dinv, NN);

    // --- layer 1: h1 = x @ W1 (100000x128 @ 128x32) ---
    {
        int tiles = (NN / 16) * (32 / 16);
        k_wmma_gemm<<<blocks(tiles, WPB), T, 0, stream>>>(x, W1, h1, nullptr,
                                                          NN, 32, 128, 0);
    }
    k_zero   <<<blocks((long long)NN * 32, T), T, 0, stream>>>(x1, (long long)NN * 32);
    k_scatter<<<blocks((long long)E << 5, T), T, 0, stream>>>(h1, src, dst, dinv, x1, E, 5);
    k_selfloop_bias_relu<<<blocks((long long)NN << 5, T), T, 0, stream>>>(
        h1, dinv, x1, b1, x1, NN, 5);

    // --- layer 2: h2 = x1 @ W2 (100000x32 @ 32x64) ---
    {
        int tiles = (NN / 16) * (64 / 16);
        k_wmma_gemm<<<blocks(tiles, WPB), T, 0, stream>>>(x1, W2, h2, nullptr,
                                                          NN, 64, 32, 0);
    }
    k_zero   <<<blocks((long long)NN * 64, T), T, 0, stream>>>(x2, (long long)NN * 64);
    k_scatter<<<blocks((long long)E << 6, T), T, 0, stream>>>(h2, src, dst, dinv, x2, E, 6);
    k_selfloop_bias_relu<<<blocks((long long)NN << 6, T), T, 0, stream>>>(
        h2, dinv, x2, b2, x2, NN, 6);

    // --- global mean pool over JK-concat [x1 | x2] -> [NG, 96] ---
    k_zero     <<<blocks(NG * 96 + NG, T), T, 0, stream>>>(sums, NG * 96 + NG); // sums + cnt
    k_pool_accum<<<blocks((long long)NN * 96, T), T, 0, stream>>>(x1, x2, batch, sums, cnt, NN);
    k_pool_final<<<blocks(NG * 96, T), T, 0, stream>>>(sums, cnt, NG * 96);

    // --- head: hidden = relu(pooled @ fW1 + fb1) (512x96 @ 96x128) ---
    {
        int tiles = (NG / 16) * (128 / 16);
        k_wmma_gemm<<<blocks(tiles, WPB), T, 0, stream>>>(sums, fW1, hidden, fb1,
                                                          NG, 128, 96, 1);
    }
    // --- out = hidden @ fW2 + fb2 ---
    k_head2<<<blocks(NG, T), T, 0, stream>>>(hidden, fW2, fb2, out, NG);
}